// OffsetHead_81423989997656
// MI455X (gfx1250) — compile-verified
//
#include <hip/hip_runtime.h>
#include <hip/hip_bf16.h>

// ---------------------------------------------------------------------------
// MI455X (gfx1250) implementation of the sparse-voxelize pipeline.
//   offsets = feats @ W + b           -> f32 WMMA 16x16x4, K=128
//   keys    = injective hash          -> compact 35-bit monotone key
//   unique  = 5-pass stable radix sort (wave32 ballot ranking)
//   segsum  = one wave per segment, contiguous after sort, no atomics
// Bandwidth-bound problem (~0.7 GB traffic @ 23.3 TB/s); WMMA used for the
// only matrix op, f32 precision preserved (results feed int truncation).
// This revision: B operand (W) staged in LDS pre-padded + transposed so the
// inner WMMA loop does one unconditional ds_load_b64 per step (no saveexec).
// ---------------------------------------------------------------------------

typedef float v2f __attribute__((ext_vector_type(2)));
typedef float v4f __attribute__((ext_vector_type(4)));
typedef float v8f __attribute__((ext_vector_type(8)));

#define LATENT 128

// ---------------------------------------------------------------- zero d_out
__global__ void zero_words_kernel(unsigned* __restrict__ p, long long nwords) {
    long long i = (long long)blockIdx.x * blockDim.x + threadIdx.x;
    long long stride = (long long)gridDim.x * blockDim.x;
    for (; i < nwords; i += stride) p[i] = 0u;
}

// ------------------------------------------------------- offsets GEMM (WMMA)
// One wave computes a 16x16 f32 tile of feats[16,128] x W[128,16-padded].
// A layout (16x4 f32): lanes 0-15 row M=lane, VGPR{0,1}=K{0,1};
//                      lanes 16-31 row M=lane-16, VGPR{0,1}=K{2,3}.
// B layout (4x16 f32): mirrored: v0 holds K=(0|2), v1 holds K=(1|3), col N=lane&15.
// D layout: VGPR r -> M = r + (lane>=16 ? 8 : 0), N = lane&15.
// W is staged in LDS transposed+zero-padded: shWt[n*128 + k], n in [0,16).
__global__ void offsets_wmma_kernel(const float* __restrict__ feats,
                                    const float* __restrict__ W,
                                    const float* __restrict__ bias,
                                    float* __restrict__ out_off, int N) {
    __shared__ float shWt[16 * LATENT];   // column-major, zero-padded to 16 cols
    __shared__ float shb[16];             // zero-padded bias
    int t = threadIdx.x;

    for (int k = t; k < 16 * LATENT; k += blockDim.x) shWt[k] = 0.0f;
    if (t < 16) shb[t] = (t < 3) ? bias[t] : 0.0f;
    __syncthreads();
    for (int k = t; k < LATENT * 3; k += blockDim.x) {
        int kk = k / 3, n = k % 3;
        shWt[n * LATENT + kk] = W[k];
    }
    __syncthreads();

    int wave = t >> 5;
    int lane = t & 31;
    int rowBase = (blockIdx.x * 8 + wave) * 16;
    if (rowBase >= N) return;                 // wave-uniform, EXEC stays all-1s

    int mrow = rowBase + (lane & 15);
    int koff = (lane >> 4) * 2;               // 0 for lanes 0-15, 2 for 16-31
    int n    = lane & 15;                     // output column (only 0..2 valid)

    v8f c = {0.f, 0.f, 0.f, 0.f, 0.f, 0.f, 0.f, 0.f};
    const float* arow = feats + (size_t)mrow * LATENT + koff;
    const float* brow = shWt + n * LATENT + koff;

    for (int kk = 0; kk < LATENT; kk += 4) {
        v2f a  = *(const v2f*)(arow + kk);    // global_load_b64, imm offset
        v2f bm = *(const v2f*)(brow + kk);    // ds_load_b64, imm offset
        c = __builtin_amdgcn_wmma_f32_16x16x4_f32(
                /*neg_a=*/false, a, /*neg_b=*/false, bm,
                /*c_mod=*/(short)0, c, /*reuse_a=*/false, /*reuse_b=*/false);
    }

    float bb = shb[n];
    int mtop = rowBase + ((lane >> 4) << 3);
#pragma unroll
    for (int r = 0; r < 8; r++) {
        int m = mtop + r;
        if (n < 3 && m < N) out_off[(size_t)m * 3 + n] = c[r] + bb;
    }
}

// ----------------------------------------------------- hash keys + identity
// Reference key = (((b+512)*2048 + x+512)*2048 + y+512)*2048 + z+512.
// Subtract 512*2^33 (b in [0,4)) -> monotone compact key < 2^35 (5x8-bit sort).
__global__ void make_keys_kernel(const float* __restrict__ offs,
                                 const int* __restrict__ coords,
                                 int* __restrict__ ncrd,
                                 unsigned long long* __restrict__ keys,
                                 int* __restrict__ idx, int N) {
    int i = blockIdx.x * blockDim.x + threadIdx.x;
    if (i >= N) return;
    int ox = (int)offs[(size_t)i * 3 + 0];    // trunc toward zero == astype(int32)
    int oy = (int)offs[(size_t)i * 3 + 1];
    int oz = (int)offs[(size_t)i * 3 + 2];
    int b = coords[(size_t)i * 4 + 0];
    int x = coords[(size_t)i * 4 + 1] + ox;
    int y = coords[(size_t)i * 4 + 2] + oy;
    int z = coords[(size_t)i * 4 + 3] + oz;
    ncrd[(size_t)i * 4 + 0] = b;
    ncrd[(size_t)i * 4 + 1] = x;
    ncrd[(size_t)i * 4 + 2] = y;
    ncrd[(size_t)i * 4 + 3] = z;
    unsigned long long k =
        ((((unsigned long long)(b + 512) * 2048ull + (unsigned long long)(x + 512)) * 2048ull
          + (unsigned long long)(y + 512)) * 2048ull + (unsigned long long)(z + 512))
        - (512ull << 33);
    keys[i] = k;
    idx[i]  = i;
}

// ------------------------------------------------------------- radix: hist
__global__ void radix_hist_kernel(const unsigned long long* __restrict__ keys,
                                  unsigned* __restrict__ hist,
                                  int N, int numBlocks, int shift) {
    __shared__ unsigned sh[256];
    int t = threadIdx.x;
    sh[t] = 0u;
    __syncthreads();
    int i = blockIdx.x * 256 + t;
    if (i < N) {
        unsigned d = (unsigned)((keys[i] >> shift) & 0xFFull);
        atomicAdd(&sh[d], 1u);
    }
    __syncthreads();
    hist[(size_t)t * numBlocks + blockIdx.x] = sh[t];
}

// --------------------------------------------------- scan (3-kernel, uint)
__global__ void scan1_kernel(unsigned* __restrict__ data,
                             unsigned* __restrict__ aux, int n) {
    __shared__ unsigned sh[256];
    int t = threadIdx.x;
    int i = blockIdx.x * 256 + t;
    unsigned v = (i < n) ? data[i] : 0u;
    sh[t] = v;
    __syncthreads();
    for (int off = 1; off < 256; off <<= 1) {
        unsigned x = (t >= off) ? sh[t - off] : 0u;
        __syncthreads();
        sh[t] += x;
        __syncthreads();
    }
    unsigned incl = sh[t];
    if (i < n) data[i] = incl - v;            // exclusive within block
    if (t == 255) aux[blockIdx.x] = incl;     // block total
}

__global__ void scan_aux_kernel(unsigned* __restrict__ aux, int m) {
    __shared__ unsigned sh[256];
    __shared__ unsigned carry;
    int t = threadIdx.x;
    if (t == 0) carry = 0u;
    __syncthreads();
    for (int base = 0; base < m; base += 256) {
        int i = base + t;
        unsigned v = (i < m) ? aux[i] : 0u;
        sh[t] = v;
        __syncthreads();
        for (int off = 1; off < 256; off <<= 1) {
            unsigned x = (t >= off) ? sh[t - off] : 0u;
            __syncthreads();
            sh[t] += x;
            __syncthreads();
        }
        unsigned incl = sh[t];
        unsigned c = carry;
        if (i < m) aux[i] = incl - v + c;
        __syncthreads();
        if (t == 255) carry = c + incl;
        __syncthreads();
    }
}

__global__ void scan2_kernel(unsigned* __restrict__ data,
                             const unsigned* __restrict__ aux, int n) {
    int i = blockIdx.x * 256 + threadIdx.x;
    if (i < n) data[i] += aux[blockIdx.x];
}

// --------------------------------------------- radix: stable scatter (wave32)
// Stable rank via 8 rounds of wave32 ballot match-mask + per-wave histograms.
__global__ void radix_scatter_kernel(const unsigned long long* __restrict__ kin,
                                     const int* __restrict__ iin,
                                     unsigned long long* __restrict__ kout,
                                     int* __restrict__ iout,
                                     const unsigned* __restrict__ hist,
                                     int N, int numBlocks, int shift) {
    __shared__ unsigned whist[8 * 256];
    int t = threadIdx.x;
    int wave = t >> 5;
    int lane = t & 31;
    for (int k = t; k < 8 * 256; k += 256) whist[k] = 0u;
    __syncthreads();

    int i = blockIdx.x * 256 + t;
    bool valid = (i < N);
    unsigned long long key = valid ? kin[i] : 0ull;
    unsigned digit = valid ? (unsigned)((key >> shift) & 0xFFull) : 0u;

    unsigned validmask = (unsigned)__ballot(valid ? 1 : 0);
    unsigned m = 0xFFFFFFFFu;
#pragma unroll
    for (int b = 0; b < 8; b++) {
        unsigned bb = (unsigned)__ballot((int)((digit >> b) & 1u));
        m &= ((digit >> b) & 1u) ? bb : ~bb;
    }
    m &= validmask;

    unsigned lt = (lane == 31) ? 0x7FFFFFFFu : ((1u << lane) - 1u);
    unsigned lane_rank = (unsigned)__popc(m & lt);
    bool leader = valid && ((m & lt) == 0u);
    if (leader) whist[wave * 256 + digit] = (unsigned)__popc(m);
    __syncthreads();

    if (valid) {
        unsigned before = 0u;
        for (int w = 0; w < wave; w++) before += whist[w * 256 + digit];
        unsigned dst = hist[(size_t)digit * numBlocks + blockIdx.x] + before + lane_rank;
        kout[dst] = key;
        iout[dst] = iin[i];
    }
}

// ----------------------------------------------------------- segment heads
__global__ void head_kernel(const unsigned long long* __restrict__ ksort,
                            unsigned* __restrict__ heads, int N) {
    int i = blockIdx.x * blockDim.x + threadIdx.x;
    if (i >= N) return;
    heads[i] = (i == 0 || ksort[i] != ksort[i - 1]) ? 1u : 0u;
}

// heads[] has been exclusive-scanned in place before this runs.
__global__ void assign_kernel(const unsigned long long* __restrict__ ksort,
                              const int* __restrict__ isort,
                              const unsigned* __restrict__ excl,
                              int* __restrict__ seg_start,
                              long long* __restrict__ out_inv,
                              int* __restrict__ numU, int N) {
    int i = blockIdx.x * blockDim.x + threadIdx.x;
    if (i >= N) return;
    bool head = (i == 0 || ksort[i] != ksort[i - 1]);
    int s = (int)excl[i] + (head ? 0 : -1);
    out_inv[isort[i]] = (long long)s;
    if (head) seg_start[s] = i;
    if (i == N - 1) *numU = s + 1;
}

// ------------------------------------------- per-segment mean (wave/segment)
__global__ void seg_reduce_kernel(const float* __restrict__ feats,
                                  const int* __restrict__ ncrd,
                                  const int* __restrict__ isort,
                                  const int* __restrict__ seg_start,
                                  const int* __restrict__ numU,
                                  float* __restrict__ out_feats,
                                  int* __restrict__ out_coords, int N) {
    int wave = threadIdx.x >> 5;
    int lane = threadIdx.x & 31;
    int s = blockIdx.x * 8 + wave;
    int U = *numU;
    if (s >= U) return;
    int start = seg_start[s];
    int end = (s + 1 < U) ? seg_start[s + 1] : N;

    v4f acc = {0.f, 0.f, 0.f, 0.f};
    float csum = 0.f;
    for (int p = start; p < end; p++) {
        int j = isort[p];
        acc += *(const v4f*)(feats + (size_t)j * LATENT + lane * 4);
        if (lane < 4) csum += (float)ncrd[(size_t)j * 4 + lane];
    }
    float cnt = (float)(end - start);
    float inv = 1.0f / cnt;
    v4f r = acc * inv;
    *(v4f*)(out_feats + (size_t)s * LATENT + lane * 4) = r;
    if (lane < 4) out_coords[(size_t)s * 4 + lane] = (int)(csum / cnt);
}

// ---------------------------------------------------------------------------
extern "C" void kernel_launch(void* const* d_in, const int* in_sizes, int n_in,
                              void* d_out, int out_size, void* d_ws, size_t ws_size,
                              hipStream_t stream) {
    const float* feats = (const float*)d_in[0];
    const int*   coords = (const int*)d_in[1];
    const float* W = (const float*)d_in[2];
    const float* bias = (const float*)d_in[3];
    int N = in_sizes[0] / LATENT;

    // d_out layout: offsets f32[3N] | out_coords i32[4N] | out_feats f32[128N] | inv i64[N]
    float*     out_off    = (float*)d_out;
    int*       out_coords = (int*)((char*)d_out + (size_t)3 * N * 4);
    float*     out_feats  = (float*)((char*)d_out + (size_t)7 * N * 4);
    long long* out_inv    = (long long*)((char*)d_out + (size_t)135 * N * 4);

    // workspace carve-out (256B aligned regions, ~21 MB total for N=400000)
    char* w = (char*)d_ws;
    auto take = [&](size_t bytes) -> char* {
        char* p = w;
        w += (bytes + 255) & ~(size_t)255;
        return p;
    };
    int numBlocks = (N + 255) / 256;
    unsigned long long* keysA = (unsigned long long*)take(8ull * N);
    unsigned long long* keysB = (unsigned long long*)take(8ull * N);
    int* idxA = (int*)take(4ull * N);
    int* idxB = (int*)take(4ull * N);
    unsigned* hist = (unsigned*)take(4ull * 256 * numBlocks);
    unsigned* aux  = (unsigned*)take(4ull * numBlocks);
    unsigned* segw = (unsigned*)take(4ull * N);
    int* seg_start = (int*)take(4ull * N);
    int* ncrd = (int*)take(16ull * N);
    int* numU = (int*)take(256);

    // 1) zero output (covers padded tail segments of unique())
    long long zwords = (long long)137 * N;
    int zgrid = (int)((zwords + 255) / 256);
    if (zgrid > 65535) zgrid = 65535;
    zero_words_kernel<<<zgrid, 256, 0, stream>>>((unsigned*)d_out, zwords);

    // 2) offsets GEMM via f32 WMMA
    offsets_wmma_kernel<<<(N + 127) / 128, 256, 0, stream>>>(feats, W, bias, out_off, N);

    // 3) hash
    make_keys_kernel<<<numBlocks, 256, 0, stream>>>(out_off, coords, ncrd, keysA, idxA, N);

    // 4) 5-pass stable LSD radix sort over 40 bits (key < 2^35)
    unsigned long long* kin = keysA; unsigned long long* kout = keysB;
    int* iin = idxA; int* iout = idxB;
    int nh = 256 * numBlocks;
    for (int pass = 0; pass < 5; pass++) {
        int shift = 8 * pass;
        radix_hist_kernel<<<numBlocks, 256, 0, stream>>>(kin, hist, N, numBlocks, shift);
        scan1_kernel<<<numBlocks, 256, 0, stream>>>(hist, aux, nh);
        scan_aux_kernel<<<1, 256, 0, stream>>>(aux, numBlocks);
        scan2_kernel<<<numBlocks, 256, 0, stream>>>(hist, aux, nh);
        radix_scatter_kernel<<<numBlocks, 256, 0, stream>>>(kin, iin, kout, iout,
                                                            hist, N, numBlocks, shift);
        unsigned long long* tk = kin; kin = kout; kout = tk;
        int* ti = iin; iin = iout; iout = ti;
    }
    // sorted pairs now in (kin, iin)

    // 5) segment ids = rank among sorted-unique keys
    head_kernel<<<numBlocks, 256, 0, stream>>>(kin, segw, N);
    scan1_kernel<<<numBlocks, 256, 0, stream>>>(segw, aux, N);
    scan_aux_kernel<<<1, 256, 0, stream>>>(aux, numBlocks);
    scan2_kernel<<<numBlocks, 256, 0, stream>>>(segw, aux, N);
    assign_kernel<<<numBlocks, 256, 0, stream>>>(kin, iin, segw, seg_start,
                                                 out_inv, numU, N);

    // 6) per-segment means (contiguous after sort -> no atomics)
    seg_reduce_kernel<<<(N + 7) / 8, 256, 0, stream>>>(feats, ncrd, iin, seg_start,
                                                       numU, out_feats, out_coords, N);
}